// ReorgLayer_7490422964738
// MI455X (gfx1250) — compile-verified
//
#include <hip/hip_runtime.h>

// Space-to-depth (YOLOv2 reorg), stride = 2, fp32.
//   out[b, i*4 + l*2 + m, oy, ox] = x[b, i, 2*oy + l, 2*ox + m]
// x: (32, 64, 256, 256) contiguous -> out: (32, 256, 128, 128) contiguous.
//
// Pure bandwidth problem: 1 GiB total traffic, 0 FLOPs -> ~46 us floor at
// 23.3 TB/s. Strategy: each thread consumes 8 consecutive input floats
// (two b128 loads); deinterleave even/odd lanes into two float4s that are
// each a contiguous 4-wide run of one output channel row (two b128 stores).
// All accesses 16B-aligned and vectorized; non-temporal hints since the
// stream is one-touch and exceeds the 192 MB L2.

typedef __attribute__((ext_vector_type(4))) float v4f;

__global__ __launch_bounds__(256) void reorg_s2_b128_kernel(
    const float* __restrict__ in, float* __restrict__ out, unsigned nThreads) {
  unsigned t = blockIdx.x * blockDim.x + threadIdx.x;
  if (t >= nThreads) return;

  // Input is fully contiguous: thread t owns floats [8t, 8t+8).
  const v4f* __restrict__ ip = (const v4f*)(in + (size_t)t * 8u);
  v4f v0 = __builtin_nontemporal_load(ip);        // x_ = 8q+0 .. 8q+3
  v4f v1 = __builtin_nontemporal_load(ip + 1);    // x_ = 8q+4 .. 8q+7

  // Decompose t: 32 eight-float segments per 256-wide input row.
  unsigned seg = t & 31u;        // q: segment within input row -> ox0 = 4q
  unsigned row = t >> 5;         // global input row = (b*64 + i)*256 + y
  unsigned y   = row & 255u;
  unsigned bc  = row >> 8;       // b*64 + i
  unsigned i   = bc & 63u;
  unsigned b   = bc >> 6;

  unsigned oy = y >> 1;
  unsigned l  = y & 1u;
  unsigned c0 = (i << 2) + (l << 1);           // channel for m = 0

  // out offset: ((b*256 + c)*128 + oy)*128 + ox0   (fits in 32 bits: < 2^27)
  unsigned out0 = ((((b << 8) + c0) << 7) + oy) * 128u + (seg << 2);

  // m=0 channel: even elements; m=1 channel: odd elements.
  v4f w0 = { v0.x, v0.z, v1.x, v1.z };
  v4f w1 = { v0.y, v0.w, v1.y, v1.w };

  __builtin_nontemporal_store(w0, (v4f*)(out + (size_t)out0));
  __builtin_nontemporal_store(w1, (v4f*)(out + (size_t)out0 + 16384u)); // +1 channel = 128*128
}

extern "C" void kernel_launch(void* const* d_in, const int* in_sizes, int n_in,
                              void* d_out, int out_size, void* d_ws, size_t ws_size,
                              hipStream_t stream) {
  (void)n_in; (void)d_ws; (void)ws_size; (void)out_size;
  const float* x = (const float*)d_in[0];
  float* out = (float*)d_out;

  // 32*64*256*256 = 134,217,728 elements; 8 per thread.
  unsigned nThreads = (unsigned)((size_t)in_sizes[0] / 8u);
  unsigned block = 256u;                       // 8 wave32s
  unsigned grid  = (nThreads + block - 1u) / block;
  reorg_s2_b128_kernel<<<grid, block, 0, stream>>>(x, out, nThreads);
}